// CarNet_58669253263511
// MI455X (gfx1250) — compile-verified
//
#include <hip/hip_runtime.h>
#include <hip/hip_bf16.h>
#include <math.h>

// Shapes from the reference
#define B_ 8
#define L_ 64
#define F_ 64
#define R_ 1024
#define T_ 1024

typedef __attribute__((ext_vector_type(2))) float v2f;
typedef __attribute__((ext_vector_type(8))) float v8f;

// ---------------------------------------------------------------------------
// Kernel 1: arg1[b,f,r] = sum_l x[b,l,f,r] * w[b,l]
// 128 MB of x streamed once, float4 per thread, weights in LDS.
// grid = B*F*R/4/256 = 512 blocks, 256 threads. Each block stays in one batch
// (16384 float4 per batch, 256 | 16384).
// ---------------------------------------------------------------------------
__global__ __launch_bounds__(256) void carnet_wsum_kernel(
    const float* __restrict__ x, const float* __restrict__ w,
    float* __restrict__ arg1) {
  __shared__ float wl[L_];
  const int idx = blockIdx.x * 256 + threadIdx.x;     // [0, 131072)
  const int b   = idx >> 14;                          // / 16384 (= F*R/4)
  const int rem = idx & 16383;                        // float4 index in [F*R/4)

  if (threadIdx.x < L_) wl[threadIdx.x] = w[b * L_ + threadIdx.x];
  __syncthreads();

  const float4* __restrict__ xp =
      reinterpret_cast<const float4*>(x) + (size_t)b * L_ * 16384 + rem;

  float4 acc = make_float4(0.f, 0.f, 0.f, 0.f);
#pragma unroll 8
  for (int l = 0; l < L_; ++l) {
    const float4 v = xp[(size_t)l * 16384];
    const float ww = wl[l];
    acc.x = fmaf(v.x, ww, acc.x);
    acc.y = fmaf(v.y, ww, acc.y);
    acc.z = fmaf(v.z, ww, acc.z);
    acc.w = fmaf(v.w, ww, acc.w);
  }
  reinterpret_cast<float4*>(arg1)[(size_t)b * 16384 + rem] = acc;
}

// ---------------------------------------------------------------------------
// Kernel 2: out[b,f,t] = sum_r arg1[b,f,r] * W[t,r]   (fp32 WMMA 16x16x4)
// grid = (T/64, B), block = 256 threads (8 waves).
// Block computes the full 64(F) x 64(T) tile for batch b, T-tile blockIdx.x.
// A and B chunks (64 x 64) staged in LDS, row stride 68 floats:
//   - 16B-aligned rows (68*4 = 272 = 17*16)
//   - fragment ds_load_b64: lanes m=0..15 hit banks 4m+k (mod 64), all
//     distinct; lanes 16..31 hit 4m+k+2 -> disjoint. Conflict-free.
// Fragment layout for V_WMMA_F32_16X16X4_F32 (ISA 7.12.2):
//   A (16x4): v0 = K0 (lanes 0-15) / K2 (lanes 16-31), v1 = K1 / K3
//   B (4x16): v0 = K0-row / K2-row, v1 = K1-row / K3-row  -> same addressing
//   C/D:      vj, lanes 0-15: M=j, N=lane; lanes 16-31: M=j+8, N=lane-16
// ---------------------------------------------------------------------------
#define LDA 68

__global__ __launch_bounds__(256) void carnet_gemm_kernel(
    const float* __restrict__ arg1, const float* __restrict__ cw,
    float* __restrict__ out) {
  __shared__ float As[64 * LDA];
  __shared__ float Bs[64 * LDA];

  const int b  = blockIdx.y;
  const int t0 = blockIdx.x * 64;

  const int tid  = threadIdx.x;
  const int lane = tid & 31;
  const int wave = tid >> 5;

  // Cooperative-load coordinates: 16 threads per row, float4 columns.
  const int lrow = tid >> 4;          // 0..15
  const int lcol = (tid & 15) << 2;   // 0,4,...,60

  // Wave owns subtiles s = 2*wave and 2*wave+1 (same fTile, adjacent tTiles).
  const int fTile = wave >> 1;        // 0..3
  const int tT0   = (wave & 1) << 1;  // 0 or 2
  const int tT1   = tT0 + 1;

  const int m    = lane & 15;
  const int koff = (lane >> 4) << 1;  // 0 for lanes 0-15, 2 for lanes 16-31

  const float* __restrict__ Aab = &As[(fTile * 16 + m) * LDA + koff];
  const float* __restrict__ Bb0 = &Bs[(tT0   * 16 + m) * LDA + koff];
  const float* __restrict__ Bb1 = &Bs[(tT1   * 16 + m) * LDA + koff];

  const float* __restrict__ Ag = arg1 + (size_t)b * F_ * R_;
  const float* __restrict__ Bg = cw + (size_t)t0 * R_;

  v8f acc0 = {};
  v8f acc1 = {};

  for (int kbase = 0; kbase < R_; kbase += 64) {
    // Stage A (arg1[b, 0..63, kbase..kbase+63]) and B (W[t0..t0+63, kbase..])
#pragma unroll
    for (int p = 0; p < 4; ++p) {
      const int row = lrow + p * 16;
      const float4 va = *reinterpret_cast<const float4*>(
          Ag + (size_t)row * R_ + kbase + lcol);
      const float4 vb = *reinterpret_cast<const float4*>(
          Bg + (size_t)row * R_ + kbase + lcol);
      *reinterpret_cast<float4*>(&As[row * LDA + lcol]) = va;
      *reinterpret_cast<float4*>(&Bs[row * LDA + lcol]) = vb;
    }
    __syncthreads();

#pragma unroll
    for (int k = 0; k < 64; k += 4) {
      const v2f a  = *reinterpret_cast<const v2f*>(Aab + k);
      const v2f b0 = *reinterpret_cast<const v2f*>(Bb0 + k);
      const v2f b1 = *reinterpret_cast<const v2f*>(Bb1 + k);
      acc0 = __builtin_amdgcn_wmma_f32_16x16x4_f32(
          false, a, false, b0, (short)0, acc0, false, false);
      acc1 = __builtin_amdgcn_wmma_f32_16x16x4_f32(
          false, a, false, b1, (short)0, acc1, false, false);
    }
    __syncthreads();
  }

  // Write back both 16x16 tiles.
  const int n    = lane & 15;
  const int half = (lane >> 4) << 3;  // +0 or +8 rows
#pragma unroll
  for (int j = 0; j < 8; ++j) {
    const int f = fTile * 16 + half + j;
    out[((size_t)b * F_ + f) * T_ + t0 + tT0 * 16 + n] = acc0[j];
    out[((size_t)b * F_ + f) * T_ + t0 + tT1 * 16 + n] = acc1[j];
  }
}

// ---------------------------------------------------------------------------
// Kernel 3: entropy[b] = (ln S - (sum w ln w)/S) / ln(L),  max_w[b]
// 8 blocks x 64 threads, LDS tree reduction.
// ---------------------------------------------------------------------------
__global__ __launch_bounds__(64) void carnet_stats_kernel(
    const float* __restrict__ w, float* __restrict__ entropy,
    float* __restrict__ maxw) {
  __shared__ float sS[L_], sWL[L_], sM[L_];
  const int b = blockIdx.x;
  const int t = threadIdx.x;

  const float v = w[b * L_ + t];
  sS[t]  = v;
  sWL[t] = v * logf(v);
  sM[t]  = v;
  __syncthreads();

  for (int off = 32; off > 0; off >>= 1) {
    if (t < off) {
      sS[t]  += sS[t + off];
      sWL[t] += sWL[t + off];
      sM[t]   = fmaxf(sM[t], sM[t + off]);
    }
    __syncthreads();
  }
  if (t == 0) {
    const float S = sS[0];
    entropy[b] = (logf(S) - sWL[0] / S) / logf((float)L_);
    maxw[b] = sM[0];
  }
}

// ---------------------------------------------------------------------------
extern "C" void kernel_launch(void* const* d_in, const int* in_sizes, int n_in,
                              void* d_out, int out_size, void* d_ws,
                              size_t ws_size, hipStream_t stream) {
  const float* x  = (const float*)d_in[0];   // [B,L,F,R]
  const float* w  = (const float*)d_in[1];   // [B,L]
  const float* cw = (const float*)d_in[2];   // [T,R]

  float* out  = (float*)d_out;               // [B,F,T] + entropy[B] + max_w[B]
  float* arg1 = (float*)d_ws;                // [B,F,R] scratch (2 MB)

  // K1: 131072 threads, one float4 each.
  carnet_wsum_kernel<<<512, 256, 0, stream>>>(x, w, arg1);

  // K2: (T/64) x B blocks of 8 waves.
  carnet_gemm_kernel<<<dim3(T_ / 64, B_), 256, 0, stream>>>(arg1, cw, out);

  // K3: per-batch entropy and max.
  float* ent = out + (size_t)B_ * F_ * T_;   // offset 524288
  float* mx  = ent + B_;                     // offset 524296
  carnet_stats_kernel<<<B_, L_, 0, stream>>>(w, ent, mx);
}